// Update_entity_50689204027759
// MI455X (gfx1250) — compile-verified
//
#include <hip/hip_runtime.h>
#include <math.h>
#include <stdint.h>

typedef __attribute__((ext_vector_type(2))) float v2f;
typedef __attribute__((ext_vector_type(8))) float v8f;
typedef __attribute__((ext_vector_type(4))) int v4i;
typedef __attribute__((address_space(1))) v4i gv4i;   // global
typedef __attribute__((address_space(3))) v4i lv4i;   // LDS

#define BB 8192
#define EE 64
#define DD 128
#define PP 4096
#define RS 132   // padded LDS row stride for h/k (conflict-free A-fragment loads)

#if __has_builtin(__builtin_amdgcn_global_load_async_to_lds_b128)
#define HAVE_ASYNC_LDS 1
#endif

__device__ __forceinline__ void stage16(const float* __restrict__ g, float* l) {
#if defined(HAVE_ASYNC_LDS)
    __builtin_amdgcn_global_load_async_to_lds_b128(
        (gv4i*)(uintptr_t)g, (lv4i*)(uintptr_t)l, 0, 0);
#else
    *(float4*)l = *(const float4*)g;
#endif
}

__device__ __forceinline__ void stage_wait() {
#if defined(HAVE_ASYNC_LDS)
#if __has_builtin(__builtin_amdgcn_s_wait_asynccnt)
    __builtin_amdgcn_s_wait_asynccnt(0);
#else
    asm volatile("s_wait_asynccnt 0x0" ::: "memory");
#endif
#endif
}

// ---------------------------------------------------------------------------
// sW[p][f] = sum_d encoded_sents[p][d] * W[d][f]
// ---------------------------------------------------------------------------
__global__ __launch_bounds__(128) void sw_kernel(const float* __restrict__ es,
                                                 const float* __restrict__ W,
                                                 float* __restrict__ sW) {
    __shared__ float sh_s[DD];
    const int p = blockIdx.x;
    const int f = threadIdx.x;
    sh_s[f] = es[(size_t)p * DD + f];
    __syncthreads();
    float acc = 0.0f;
#pragma unroll 8
    for (int d = 0; d < DD; ++d) acc += sh_s[d] * W[d * DD + f];
    sW[(size_t)p * DD + f] = acc;
}

// ---------------------------------------------------------------------------
// Main fused kernel. One block = one p (all 64 e-rows), 128 threads = 4 waves,
// wave w owns e-rows 16w..16w+15. U/V column tiles staged per-tile in LDS and
// shared by all 4 waves; h/k rows staged via async global->LDS with padded
// row stride (132) so A-fragment ds loads are bank-conflict-free.
// WMMA: V_WMMA_F32_16X16X4_F32, two per k-step (h@U and k@V into one acc).
// ---------------------------------------------------------------------------
__global__ __launch_bounds__(128) void update_kernel(
    const float* __restrict__ hiddens, const float* __restrict__ keys,
    const float* __restrict__ es, const float* __restrict__ U,
    const float* __restrict__ V, const float* __restrict__ sW,
    const int* __restrict__ indices, float* __restrict__ out) {
    extern __shared__ float smem[];
    float* sh_h  = smem;                  // 64*RS = 8448
    float* sh_k  = sh_h + 64 * RS;        // 8448
    float* sh_bu = sh_k + 64 * RS;        // 128*16 = 2048
    float* sh_bv = sh_bu + DD * 16;       // 2048
    float* sh_s  = sh_bv + DD * 16;       // 128
    float* sh_w  = sh_s + DD;             // 128
    float* sh_g  = sh_w + DD;             // 64

    const int p    = blockIdx.x;
    const int tid  = threadIdx.x;         // 0..127
    const int lane = tid & 31;
    const int wv   = tid >> 5;            // 0..3
    const int m    = lane & 15;           // tile row (A) / tile col (B,C)
    const int hi   = lane >> 4;           // lane half
    const int idx  = indices[p];

    // ---- stage h,k: 64 rows x 32 float4, padded LDS rows ----
    const float* gh = hiddens + (size_t)idx * EE * DD;
    const float* gk = keys + (size_t)idx * EE * DD;
#pragma unroll 4
    for (int i = tid; i < 2048; i += 128) {
        const int row = i >> 5, c4 = (i & 31) * 4;
        stage16(gh + i * 4, sh_h + row * RS + c4);
        stage16(gk + i * 4, sh_k + row * RS + c4);
    }
    if (tid < 32) {
        ((float4*)sh_s)[tid] = ((const float4*)(es + (size_t)p * DD))[tid];
        ((float4*)sh_w)[tid] = ((const float4*)(sW + (size_t)p * DD))[tid];
    }
    stage_wait();
    __syncthreads();

    // ---- gates: wave w handles rows 16w..16w+15, 2 lanes per row ----
    {
        const int r  = wv * 16 + m;
        const int h0 = hi * 64;
        const float* hr = sh_h + r * RS + h0;
        const float* kr = sh_k + r * RS + h0;
        const float* sr = sh_s + h0;
        float sum = 0.0f;
#pragma unroll 8
        for (int d = 0; d < 64; ++d) sum += sr[d] * (hr[d] + kr[d]);
        sum += __shfl_xor(sum, 16, 32);
        if (hi == 0) sh_g[r] = 1.0f / (1.0f + __expf(-sum));
    }
    __syncthreads();

    // per-lane, loop-invariant base pointers (immediate offsets inside the k-loop)
    const float* hp = sh_h + (wv * 16 + m) * RS + 2 * hi;
    const float* kp = sh_k + (wv * 16 + m) * RS + 2 * hi;
    const float* up = sh_bu + 32 * hi + m;   // B[k][n] tile: index 64*kk+32*hi+m
    const float* vp = sh_bv + 32 * hi + m;

    for (int t = 0; t < 8; ++t) {
        const int f0 = t * 16;
        __syncthreads();   // all waves done reading previous B tile
        // ---- stage U/V column tile [128 x 16] into LDS (512 float4 each) ----
#pragma unroll
        for (int i = tid; i < 512; i += 128) {
            const int row = i >> 2, c4 = (i & 3) * 4;
            *(float4*)(sh_bu + row * 16 + c4) =
                *(const float4*)(U + row * DD + f0 + c4);
            *(float4*)(sh_bv + row * 16 + c4) =
                *(const float4*)(V + row * DD + f0 + c4);
        }
        __syncthreads();

        v8f acc = {};
#pragma unroll 8
        for (int kk = 0; kk < 32; ++kk) {
            // A 16x4 f32: VGPR0=K{0|2}, VGPR1=K{1|3} per lane half (8B aligned)
            const v2f ah = *(const v2f*)(hp + 4 * kk);
            const v2f ak = *(const v2f*)(kp + 4 * kk);
            v2f bu, bv;
            bu.x = up[64 * kk];
            bu.y = up[64 * kk + 16];
            bv.x = vp[64 * kk];
            bv.y = vp[64 * kk + 16];
            acc = __builtin_amdgcn_wmma_f32_16x16x4_f32(
                false, ah, false, bu, (short)0, acc, false, false);
            acc = __builtin_amdgcn_wmma_f32_16x16x4_f32(
                false, ak, false, bv, (short)0, acc, false, false);
        }

        // ---- epilogue: relu(acc + sW) * gate, atomic scatter-add ----
        const float swv = sh_w[f0 + m];
        float* obase = out + ((size_t)idx * EE + wv * 16 + 8 * hi) * DD + f0 + m;
#pragma unroll
        for (int r = 0; r < 8; ++r) {
            const int e = wv * 16 + r + 8 * hi;
            float val = fmaxf(acc[r] + swv, 0.0f);
            atomicAdd(obase + (size_t)r * DD, sh_g[e] * val);
        }
    }
}

// ---------------------------------------------------------------------------
// Row-wise L2 normalization of all B*E rows: one wave per 128-float row.
// ---------------------------------------------------------------------------
__global__ __launch_bounds__(256) void norm_kernel(float* __restrict__ out) {
    const int lane = threadIdx.x & 31;
    const size_t row = (size_t)blockIdx.x * 8 + (threadIdx.x >> 5);
    float4* rp = (float4*)(out + row * DD);
    float4 v = rp[lane];
    float sq = v.x * v.x + v.y * v.y + v.z * v.z + v.w * v.w;
#pragma unroll
    for (int off = 16; off >= 1; off >>= 1) sq += __shfl_xor(sq, off, 32);
    const float r = rsqrtf(fmaxf(sq, 1e-12f));
    v.x *= r; v.y *= r; v.z *= r; v.w *= r;
    rp[lane] = v;
}

extern "C" void kernel_launch(void* const* d_in, const int* in_sizes, int n_in,
                              void* d_out, int out_size, void* d_ws, size_t ws_size,
                              hipStream_t stream) {
    const float* hiddens = (const float*)d_in[0];
    const float* keys    = (const float*)d_in[1];
    const float* es      = (const float*)d_in[2];
    const float* U       = (const float*)d_in[3];
    const float* V       = (const float*)d_in[4];
    const float* W       = (const float*)d_in[5];
    const int*   indices = (const int*)d_in[6];
    float*       out     = (float*)d_out;
    float*       sW      = (float*)d_ws;   // P*D floats = 2 MB

    (void)hipMemcpyAsync(out, hiddens, (size_t)BB * EE * DD * sizeof(float),
                         hipMemcpyDeviceToDevice, stream);

    sw_kernel<<<PP, 128, 0, stream>>>(es, W, sW);

    const size_t smem_bytes =
        (2 * 64 * RS + 2 * DD * 16 + 2 * DD + 64) * sizeof(float);  // ~85 KB
    update_kernel<<<PP, 128, smem_bytes, stream>>>(hiddens, keys, es, U, V, sW,
                                                   indices, out);

    norm_kernel<<<(BB * EE) / 8, 256, 0, stream>>>(out);
}